// MultiHeadHypergraphAttention_45449343926752
// MI455X (gfx1250) — compile-verified
//
#include <hip/hip_runtime.h>
#include <math.h>

// ---------------------------------------------------------------------------
// MultiHeadHypergraphAttention for MI455X (gfx1250, wave32)
//   Phase 1: Q/K/V projections  -> v_wmma_f32_16x16x4_f32 (exact fp32 GEMM),
//            32-row M tiles so each B fragment feeds two WMMAs
//   Phase 2: per-hyperedge 8x8x4-head attention, LDS-staged gathers with
//            per-head padding (stride 65) -> conflict-free LDS reads
//   Phase 3: scatter-add via global_atomic_add_f32 + counts
//   Phase 4: normalize + output projection -> WMMA again
// ---------------------------------------------------------------------------

typedef float v2f __attribute__((ext_vector_type(2)));
typedef float v8f __attribute__((ext_vector_type(8)));

#define N_NODES   100000
#define N_HEDGES  50000
#define EDGE      8
#define DIM       256
#define HEADS     4
#define HEAD_DIM  64

// GEMM A-tile row stride (256 -> 260: 260 mod 64 = 4, the 16 M-rows of a
// fragment land on banks 4*m apart -> conflict-free ds reads)
#define APAD 260

// attention LDS padding: per-node stride 260, per-head stride 65
//   Q read bank = (4*i + h + d) mod 64  -> all 32 lanes distinct
//   K/V read    = 4 distinct banks, each broadcast to 8 lanes
#define NPAD 260
#define HPAD 65

// workspace offsets (in floats)
#define Q_OFF   ((size_t)0)
#define K_OFF   ((size_t)N_NODES * DIM)
#define V_OFF   ((size_t)2 * N_NODES * DIM)
#define A_OFF   ((size_t)3 * N_NODES * DIM)      // accumulators
#define C_OFF   ((size_t)4 * N_NODES * DIM)      // counts (N_NODES floats)

// ---------------------------------------------------------------------------
__global__ __launch_bounds__(256)
void zero_kernel(float* __restrict__ p, int n) {
    int idx = blockIdx.x * 256 + threadIdx.x;
    int stride = gridDim.x * 256;
    for (int i = idx; i < n; i += stride) p[i] = 0.0f;
}

// ---------------------------------------------------------------------------
// Core 32x256 GEMM tile body shared by phases 1 and 4.
// ldsA: 32 x APAD staged A tile. Each wave: cols n0..n0+31, both 16-row
// halves -> 4 accumulators, 4 WMMAs per k-step off one B fragment pair.
__device__ __forceinline__
void gemm_tile_32(const float* __restrict__ ldsA,
                  const float* __restrict__ W, const float* __restrict__ bias,
                  float* __restrict__ out, int m0, int tid)
{
    const int wave  = tid >> 5;
    const int lane  = tid & 31;
    const int n0    = wave * 32;
    const int mrow  = lane & 15;          // A fragment M index
    const int khalf = (lane >> 4) << 1;   // A/B fragment K sub-offset (0 or 2)
    const int ncol  = n0 + (lane & 15);   // B/C fragment N index

    v8f acc00 = {};   // rows 0-15,  cols n0..n0+15
    v8f acc01 = {};   // rows 0-15,  cols n0+16..n0+31
    v8f acc10 = {};   // rows 16-31, cols n0..n0+15
    v8f acc11 = {};   // rows 16-31, cols n0+16..n0+31

    for (int kk = 0; kk < DIM; kk += 4) {
        v2f a0, a1;
        a0.x = ldsA[mrow * APAD + kk + khalf];
        a0.y = ldsA[mrow * APAD + kk + khalf + 1];
        a1.x = ldsA[(mrow + 16) * APAD + kk + khalf];
        a1.y = ldsA[(mrow + 16) * APAD + kk + khalf + 1];

        const float* wp = W + (size_t)(kk + khalf) * DIM;
        v2f b0, b1;
        b0.x = wp[ncol];        b0.y = wp[DIM + ncol];
        b1.x = wp[ncol + 16];   b1.y = wp[DIM + ncol + 16];

        acc00 = __builtin_amdgcn_wmma_f32_16x16x4_f32(false, a0, false, b0,
                                                      (short)0, acc00, false, false);
        acc01 = __builtin_amdgcn_wmma_f32_16x16x4_f32(false, a0, false, b1,
                                                      (short)0, acc01, false, false);
        acc10 = __builtin_amdgcn_wmma_f32_16x16x4_f32(false, a1, false, b0,
                                                      (short)0, acc10, false, false);
        acc11 = __builtin_amdgcn_wmma_f32_16x16x4_f32(false, a1, false, b1,
                                                      (short)0, acc11, false, false);
    }

    // C/D layout: VGPR r -> M = r + (lane>=16 ? 8 : 0), N = n0 + (lane&15)
    const int mbase = (lane >> 4) * 8;
    const float bia0 = bias[ncol];
    const float bia1 = bias[ncol + 16];
    #pragma unroll
    for (int r = 0; r < 8; ++r) {
        const size_t mA = (size_t)(m0 + mbase + r);
        const size_t mB = mA + 16;
        out[mA * DIM + ncol]      = acc00[r] + bia0;
        out[mA * DIM + ncol + 16] = acc01[r] + bia1;
        out[mB * DIM + ncol]      = acc10[r] + bia0;
        out[mB * DIM + ncol + 16] = acc11[r] + bia1;
    }
}

// ---------------------------------------------------------------------------
// Fused QKV projection. grid = (N_NODES/32, 3). Block: 256 threads = 8 waves.
__global__ __launch_bounds__(256)
void qkv_gemm(const float* __restrict__ x,
              const float* __restrict__ Wq, const float* __restrict__ bq,
              const float* __restrict__ Wk, const float* __restrict__ bk,
              const float* __restrict__ Wv, const float* __restrict__ bv,
              float* __restrict__ Qo, float* __restrict__ Ko,
              float* __restrict__ Vo)
{
    const int m0 = blockIdx.x * 32;

    const float* W;  const float* bias;  float* out;
    if (blockIdx.y == 0)      { W = Wq; bias = bq; out = Qo; }
    else if (blockIdx.y == 1) { W = Wk; bias = bk; out = Ko; }
    else                      { W = Wv; bias = bv; out = Vo; }

    __shared__ float ldsA[32 * APAD];   // 33.3 KB

    const int tid = threadIdx.x;
    // stage 32x256 A tile (float4 coalesced loads, padded rows)
    for (int idx = tid; idx < 32 * 64; idx += 256) {
        const int row = idx >> 6, c4 = idx & 63;
        float4 v = reinterpret_cast<const float4*>(x)[(size_t)(m0 + row) * 64 + c4];
        *reinterpret_cast<float4*>(&ldsA[row * APAD + c4 * 4]) = v;
    }
    __syncthreads();

    gemm_tile_32(ldsA, W, bias, out, m0, tid);
}

// ---------------------------------------------------------------------------
// Per-hyperedge attention. One wave per hyperedge, 2 waves per block.
// Lane mapping: i = lane>>2 (node in edge), h = lane&3 (head).
// Softmax over j is fully lane-local (8 scores in registers).
__global__ __launch_bounds__(64)
void hedge_attn(const int* __restrict__ hed,
                const float* __restrict__ Qg, const float* __restrict__ Kg,
                const float* __restrict__ Vg,
                float* __restrict__ accum, float* __restrict__ cnts)
{
    __shared__ float lds[2 * 3 * EDGE * NPAD];   // 48.75 KB

    const int tid  = threadIdx.x;
    const int wave = tid >> 5;
    const int lane = tid & 31;
    const int g    = blockIdx.x * 2 + wave;

    float* Lq = lds + wave * (3 * EDGE * NPAD);
    float* Lk = Lq + EDGE * NPAD;
    float* Lv = Lk + EDGE * NPAD;

    int nodes[EDGE];
    #pragma unroll
    for (int j = 0; j < EDGE; ++j) nodes[j] = hed[g * EDGE + j];

    // stage gathered Q/K/V. Global side: float4 coalesced. LDS side: padded
    // layout j*NPAD + h*HPAD + d (HPAD=65 breaks float4 alignment -> scalar
    // ds stores; each float4 chunk stays inside one head: 64 % 4 == 0).
    for (int idx = lane; idx < EDGE * 64; idx += 32) {
        const int j  = idx >> 6, c4 = idx & 63;
        const int h  = c4 >> 4;              // head of this 4-float chunk
        const int d0 = (c4 & 15) * 4;        // within-head dim offset
        const int dst = j * NPAD + h * HPAD + d0;
        const size_t src = (size_t)nodes[j] * 64 + c4;

        float4 q = reinterpret_cast<const float4*>(Qg)[src];
        float4 k = reinterpret_cast<const float4*>(Kg)[src];
        float4 v = reinterpret_cast<const float4*>(Vg)[src];
        Lq[dst] = q.x; Lq[dst + 1] = q.y; Lq[dst + 2] = q.z; Lq[dst + 3] = q.w;
        Lk[dst] = k.x; Lk[dst + 1] = k.y; Lk[dst + 2] = k.z; Lk[dst + 3] = k.w;
        Lv[dst] = v.x; Lv[dst + 1] = v.y; Lv[dst + 2] = v.z; Lv[dst + 3] = v.w;
    }
    __syncthreads();

    const int i    = lane >> 2;      // query node within edge
    const int h    = lane & 3;       // head
    const int base = h * HPAD;

    float s[EDGE];
    #pragma unroll
    for (int j = 0; j < EDGE; ++j) s[j] = 0.0f;

    const float* qrow = Lq + i * NPAD + base;
    for (int d = 0; d < HEAD_DIM; ++d) {
        const float qd = qrow[d];
        #pragma unroll
        for (int j = 0; j < EDGE; ++j)
            s[j] = fmaf(qd, Lk[j * NPAD + base + d], s[j]);
    }

    // scale by 1/sqrt(head_dim) = 1/8, then softmax over j (lane-local)
    float mx = -3.402823466e+38f;
    #pragma unroll
    for (int j = 0; j < EDGE; ++j) { s[j] *= 0.125f; mx = fmaxf(mx, s[j]); }
    float sum = 0.0f;
    #pragma unroll
    for (int j = 0; j < EDGE; ++j) { s[j] = __expf(s[j] - mx); sum += s[j]; }
    const float inv = 1.0f / sum;
    #pragma unroll
    for (int j = 0; j < EDGE; ++j) s[j] *= inv;

    // attended = attn @ V, scatter-add into node accumulator
    float* dst = accum + (size_t)nodes[i] * DIM + h * HEAD_DIM;
    for (int d = 0; d < HEAD_DIM; ++d) {
        float acc = 0.0f;
        #pragma unroll
        for (int j = 0; j < EDGE; ++j)
            acc = fmaf(s[j], Lv[j * NPAD + base + d], acc);
        atomicAdd(&dst[d], acc);
    }

    if (h == 0) atomicAdd(&cnts[nodes[i]], 1.0f);
}

// ---------------------------------------------------------------------------
// Normalize by counts + output projection. grid = N_NODES/32.
__global__ __launch_bounds__(256)
void out_proj(const float* __restrict__ accum, const float* __restrict__ cnts,
              const float* __restrict__ Wo, const float* __restrict__ bo,
              float* __restrict__ out)
{
    const int m0 = blockIdx.x * 32;

    __shared__ float ldsA[32 * APAD];
    __shared__ float cinv[32];

    const int tid = threadIdx.x;
    if (tid < 32) {
        const float c = cnts[m0 + tid];
        cinv[tid] = 1.0f / fmaxf(c, 1.0f);
    }
    __syncthreads();

    for (int idx = tid; idx < 32 * 64; idx += 256) {
        const int row = idx >> 6, c4 = idx & 63;
        float4 v = reinterpret_cast<const float4*>(accum)[(size_t)(m0 + row) * 64 + c4];
        const float ci = cinv[row];
        v.x *= ci; v.y *= ci; v.z *= ci; v.w *= ci;
        *reinterpret_cast<float4*>(&ldsA[row * APAD + c4 * 4]) = v;
    }
    __syncthreads();

    gemm_tile_32(ldsA, Wo, bo, out, m0, tid);
}

// ---------------------------------------------------------------------------
extern "C" void kernel_launch(void* const* d_in, const int* in_sizes, int n_in,
                              void* d_out, int out_size, void* d_ws, size_t ws_size,
                              hipStream_t stream) {
    const float* x   = (const float*)d_in[0];
    const int*   hed = (const int*)  d_in[1];
    const float* Wq  = (const float*)d_in[2];
    const float* bq  = (const float*)d_in[3];
    const float* Wk  = (const float*)d_in[4];
    const float* bk  = (const float*)d_in[5];
    const float* Wv  = (const float*)d_in[6];
    const float* bv  = (const float*)d_in[7];
    const float* Wo  = (const float*)d_in[8];
    const float* bo  = (const float*)d_in[9];

    float* ws = (float*)d_ws;
    float* Q  = ws + Q_OFF;
    float* K  = ws + K_OFF;
    float* V  = ws + V_OFF;
    float* A  = ws + A_OFF;
    float* C  = ws + C_OFF;

    // zero accumulators + counts (contiguous: N_NODES*DIM + N_NODES floats)
    const int nz = N_NODES * DIM + N_NODES;
    zero_kernel<<<2048, 256, 0, stream>>>(A, nz);

    // Q/K/V projections: grid (100000/32, 3) — exact division, EXEC all-ones
    qkv_gemm<<<dim3(N_NODES / 32, 3), 256, 0, stream>>>(
        x, Wq, bq, Wk, bk, Wv, bv, Q, K, V);

    // per-hyperedge attention + scatter: 2 hedges per block — exact division
    hedge_attn<<<N_HEDGES / 2, 64, 0, stream>>>(hed, Q, K, V, A, C);

    // normalize + output projection
    out_proj<<<N_NODES / 32, 256, 0, stream>>>(A, C, Wo, bo, (float*)d_out);
}